// Cvxpy_81174881894666
// MI455X (gfx1250) — compile-verified
//
#include <hip/hip_runtime.h>

// MI455X (gfx1250) dual-ascent solver:
//   one workgroup (256 threads, 8 wave32) per batch element
//   a[128][256] f32 staged once into LDS via GLOBAL_LOAD_ASYNC_TO_LDS_B128
//   u = a^T lam  : V_WMMA_F32_16X16X4_F32, lam broadcast into A-operand rows
//   g = a y + b  : VALU FMAs, diagonal-staggered ds_load_b128 (conflict-free)
//   sigmoid via v_exp_f32 + v_rcp_f32 (shortest serial chain per iteration)
// ~134KB LDS/block -> 2 workgroups per WGP (320KB LDS).

#define BM 128
#define BN 256
#define NITERS 100
#define STEPSZ 0.05f

typedef __attribute__((ext_vector_type(2))) float v2f;
typedef __attribute__((ext_vector_type(8))) float v8f;

__global__ __launch_bounds__(256)
void cvx_dual_ascent_kernel(const float* __restrict__ a,
                            const float* __restrict__ bvec,
                            const float* __restrict__ cvec,
                            float* __restrict__ yout,
                            int* __restrict__ status) {
  __shared__ __align__(16) float a_s[BM * BN];   // 128 KB, row-major [m][n]
  __shared__ __align__(16) float lam_s[BM];
  __shared__ __align__(16) float y_s[BN];
  __shared__ __align__(16) float u_s[BN];
  __shared__ __align__(16) float gp_s[2 * BM];

  const int tid  = (int)threadIdx.x;
  const int bidx = (int)blockIdx.x;
  const unsigned lane = (unsigned)tid & 31u;
  const unsigned wv   = (unsigned)tid >> 5;

  const float* ag = a + (size_t)bidx * (BM * BN);

  // ---- async copy a -> LDS (ASYNCcnt path). 256 lanes x 32 x 16B = 128KB,
  // each round is a 4KB fully-coalesced slab.
  {
    const unsigned lds_base =
        (unsigned)(unsigned long long)(const void*)a_s;      // wave-relative LDS offset
    const unsigned long long gbase = (unsigned long long)(const void*)ag;
#pragma unroll
    for (int k = 0; k < 32; ++k) {
      const unsigned off  = (unsigned)(k * 4096 + tid * 16);
      const unsigned ldsa = lds_base + off;
      const unsigned long long gaddr = gbase + (unsigned long long)off;
      asm volatile("global_load_async_to_lds_b128 %0, %1, off"
                   :: "v"(ldsa), "v"(gaddr) : "memory");
    }
  }

  // scalar operands while the async DMA is in flight
  const float cv = cvec[(size_t)bidx * BN + tid];
  float bv = 0.0f;
  if (tid < BM) {
    bv = bvec[(size_t)bidx * BM + tid];
    lam_s[tid] = 0.0f;
  }

  asm volatile("s_wait_asynccnt 0" ::: "memory");
  __syncthreads();

  for (int it = 0; it <= NITERS; ++it) {
    // ---------- u = a^T lam ----------
#if __has_builtin(__builtin_amdgcn_wmma_f32_16x16x4_f32)
    {
      // Each wave owns two 16-wide n-tiles; 32 chained K=4 WMMAs per tile.
      // A-operand: A[i][k] = lam[m0+k] for every row i (lane-half selects K pair).
      // B-operand: B[k][j] = a[m0+k][n0+j].
      const unsigned hi  = lane >> 4;       // 0: lanes 0-15 (K=0,1), 1: lanes 16-31 (K=2,3)
      const unsigned col = lane & 15u;
      const int n0 = (int)(wv * 32);
      const int n1 = n0 + 16;
      v8f acc0 = {0.f, 0.f, 0.f, 0.f, 0.f, 0.f, 0.f, 0.f};
      v8f acc1 = {0.f, 0.f, 0.f, 0.f, 0.f, 0.f, 0.f, 0.f};
#pragma unroll 4
      for (int m0 = 0; m0 < BM; m0 += 4) {
        const int r = m0 + (int)(hi << 1);
        v2f av; av.x = lam_s[r];                    av.y = lam_s[r + 1];
        v2f b0; b0.x = a_s[r * BN + n0 + (int)col]; b0.y = a_s[(r + 1) * BN + n0 + (int)col];
        v2f b1; b1.x = a_s[r * BN + n1 + (int)col]; b1.y = a_s[(r + 1) * BN + n1 + (int)col];
        acc0 = __builtin_amdgcn_wmma_f32_16x16x4_f32(false, av, false, b0,
                                                     (short)0, acc0, false, false);
        acc1 = __builtin_amdgcn_wmma_f32_16x16x4_f32(false, av, false, b1,
                                                     (short)0, acc1, false, false);
      }
      // C layout: VGPR0, lanes 0-15 hold (M=0, N=lane) -> u segment.
      if (lane < 16u) {
        u_s[n0 + (int)col] = acc0[0];
        u_s[n1 + (int)col] = acc1[0];
      }
    }
#else
    {
      float acc = 0.0f;
      for (int m = 0; m < BM; ++m) acc = fmaf(a_s[m * BN + tid], lam_s[m], acc);
      u_s[tid] = acc;
    }
#endif
    __syncthreads();

    // ---------- y = sigmoid(-(c + u)) = rcp(1 + exp(c + u)) ----------
    const float x  = cv + u_s[tid];
    const float yv = __builtin_amdgcn_rcpf(1.0f + __expf(x));
    if (it == NITERS) {
      yout[(size_t)bidx * BN + tid] = yv;
    }
    y_s[tid] = yv;
    __syncthreads();
    if (it == NITERS) break;   // uniform across the block

    // ---------- g = a y + b ; lam = max(lam + STEP*g, 0) ----------
    {
      // thread t: row m = t&127, half = t>>7 covers 128 of the 256 n's.
      // Diagonal stagger on the float4 chunk index keeps all 64 LDS banks busy.
      const int m    = tid & (BM - 1);
      const int half = tid >> 7;
      float sx = 0.f, sy = 0.f, sz = 0.f, sw = 0.f;
#pragma unroll 4
      for (int ci = 0; ci < 32; ++ci) {
        const int q = (half << 5) + ((ci + m) & 31);
        const float4 av4 = *(const float4*)&a_s[m * BN + (q << 2)];
        const float4 yv4 = *(const float4*)&y_s[q << 2];
        sx = fmaf(av4.x, yv4.x, sx);
        sy = fmaf(av4.y, yv4.y, sy);
        sz = fmaf(av4.z, yv4.z, sz);
        sw = fmaf(av4.w, yv4.w, sw);
      }
      gp_s[tid] = (sx + sy) + (sz + sw);
    }
    __syncthreads();
    if (tid < BM) {
      const float g = gp_s[tid] + gp_s[tid + BM] + bv;
      lam_s[tid] = fmaxf(fmaf(STEPSZ, g, lam_s[tid]), 0.0f);
    }
    __syncthreads();
  }

  if (tid == 0) status[bidx] = 0;
}

extern "C" void kernel_launch(void* const* d_in, const int* in_sizes, int n_in,
                              void* d_out, int out_size, void* d_ws, size_t ws_size,
                              hipStream_t stream) {
  (void)n_in; (void)out_size; (void)d_ws; (void)ws_size;
  const float* a = (const float*)d_in[0];
  const float* b = (const float*)d_in[1];
  const float* c = (const float*)d_in[2];
  const int B = in_sizes[0] / (BM * BN);            // 2048
  float* y = (float*)d_out;                         // [B, 256] f32
  int* status = (int*)((char*)d_out + (size_t)B * BN * sizeof(float));  // [B] i32
  cvx_dual_ascent_kernel<<<dim3((unsigned)B), dim3(256), 0, stream>>>(a, b, c, y, status);
}